// RuleNetwork_88785563943769
// MI455X (gfx1250) — compile-verified
//
#include <hip/hip_runtime.h>

// RuleNetwork forward for gfx1250 (MI455X, wave32, WMMA).
//
// Shapes: B=262144, NV=6, CV=6, NR=4, CR=64, CM1=32, CM2=16, H3=2.
// selector1 is folded into a batch-constant A1[6,4] + c1[4] (prep kernel),
// so att1 and k2 both become K=6 GEMMs against hidden — mapped onto
// V_WMMA_F32_16X16X4_F32 with 16 batch elements per wave as the N dim.
// wk2 A-operands are pre-swizzled into per-lane float4 entries so the main
// kernel gathers them with single b128 loads and no exec-divergent loads.

typedef __attribute__((ext_vector_type(2))) float v2f;
typedef __attribute__((ext_vector_type(8))) float v8f;

#define NVV 6
#define CVV 6
#define NRR 4
#define CRR 64
#define CM1C 32
#define CM2C 16

// workspace layout (floats):
//   ws[0..23]   : A1t[c*4+r]  (selector1 folded weights, scaled 1/sqrt(32))
//   ws[24..27]  : c1[r]       (selector1 folded bias)
//   ws[28..31]  : zero pad
//   ws[32..799] : wk2 A-operand lane table: float4 per (v, lane), 6*32 entries
#define WS_WK2 32

// ---------------- Phase 0: fold selector1 weights + swizzle wk2 ----------------
__global__ void rulenet_prep(const float* __restrict__ rule_emb,
                             const float* __restrict__ wq1,
                             const float* __restrict__ bq1,
                             const float* __restrict__ wk1,
                             const float* __restrict__ bk1,
                             const float* __restrict__ wk2,
                             float* __restrict__ ws) {
  int t = threadIdx.x;
  const float inv = 0.17677669529663687f; // 1/sqrt(32)
  if (t < 28) {
    int r = (t < 24) ? (t & 3) : (t - 24);
    int c = (t < 24) ? (t >> 2) : 0;
    float s = 0.f;
    for (int m = 0; m < CM1C; ++m) {
      float q1 = bq1[m];
      for (int k = 0; k < CRR; ++k)
        q1 = fmaf(rule_emb[r * CRR + k], wq1[k * CM1C + m], q1);
      float w = (t < 24) ? wk1[c * CM1C + m] : bk1[m];
      s = fmaf(w, q1, s);
    }
    ws[t] = s * inv;
  } else if (t < 32) {
    ws[t] = 0.f;
  }
  // wk2 -> A-operand lane layout: rows m (=lane%16), K=c split across halves.
  if (t < NVV * 32) {
    int v = t >> 5, lane = t & 31;
    int row = lane & 15;
    int kb = (lane >> 4) << 1;   // lanes<16: c=0,1 ; lanes>=16: c=2,3
    bool lo = lane < 16;
    float4 e;
    e.x = wk2[v * CVV * CM2C + (kb + 0) * CM2C + row];
    e.y = wk2[v * CVV * CM2C + (kb + 1) * CM2C + row];
    e.z = lo ? wk2[v * CVV * CM2C + 4 * CM2C + row] : 0.f; // K=c4 (c6 pad=0)
    e.w = lo ? wk2[v * CVV * CM2C + 5 * CM2C + row] : 0.f; // K=c5 (c7 pad=0)
    ((float4*)(ws + WS_WK2))[t] = e;
  }
}

// ---------------- Phase 1: main fused kernel ----------------
// One wave = 16 batch elements. Lane b (<16) and lane b+16 share batch b0+b.
__global__ __launch_bounds__(256) void rulenet_main(
    const float* __restrict__ hidden,
    const float* __restrict__ wq2, const float* __restrict__ bq2,
    const float* __restrict__ bk2,
    const float* __restrict__ w1,  const float* __restrict__ b1,
    const float* __restrict__ w2,  const float* __restrict__ b2,
    const float* __restrict__ ws,
    float* __restrict__ out, int batch) {
  const int lane = threadIdx.x & 31;
  const int wave = (int)((blockIdx.x * blockDim.x + threadIdx.x) >> 5);
  const int b0 = wave << 4;
  if (b0 >= batch) return;          // wave-uniform: EXEC stays all-ones below
  const bool lo = lane < 16;
  const int row = lane & 15;        // A-operand row / batch sub-index
  const int b = b0 + row;           // this lane's batch element

  // ---- load hidden[b, :, :] (36 floats, 9x float4) into registers ----
  float hid[NVV * CVV];
  {
    const float4* hp = (const float4*)(hidden + (size_t)b * (NVV * CVV));
#pragma unroll
    for (int i = 0; i < 9; ++i) {
      float4 t = hp[i];
      hid[i * 4 + 0] = t.x; hid[i * 4 + 1] = t.y;
      hid[i * 4 + 2] = t.z; hid[i * 4 + 3] = t.w;
    }
  }

  // ---- A operand for att1: A1t (rows r=0..3 valid, K = channel c) ----
  // A layout 16x4 f32: lanes 0-15 regs = K0,K1 ; lanes 16-31 regs = K2,K3.
  const int kb = (lane >> 4) << 1;          // 0 or 2
  const int rr = (row < 4) ? row : 0;       // clamp (rows 4..15 are padding)
  const bool rok = row < 4;
  // unconditional loads (all in-bounds), then register selects — no exec math
  float a1_k0 = ws[(kb + 0) * 4 + rr];
  float a1_k1 = ws[(kb + 1) * 4 + rr];
  float a1_c4 = ws[4 * 4 + rr];
  float a1_c5 = ws[5 * 4 + rr];
  float c1_0 = ws[24], c1_1 = ws[25], c1_2 = ws[26], c1_3 = ws[27];
  v2f A1a, A1b;
  A1a.x = rok ? a1_k0 : 0.f;
  A1a.y = rok ? a1_k1 : 0.f;
  A1b.x = (rok && lo) ? a1_c4 : 0.f;
  A1b.y = (rok && lo) ? a1_c5 : 0.f;

  // C operand for att1: per-rule bias c1[r] broadcast over columns
  v8f catt1;
  catt1[0] = lo ? c1_0 : 0.f; catt1[1] = lo ? c1_1 : 0.f;
  catt1[2] = lo ? c1_2 : 0.f; catt1[3] = lo ? c1_3 : 0.f;
  catt1[4] = 0.f; catt1[5] = 0.f; catt1[6] = 0.f; catt1[7] = 0.f;

  const int mbase = lo ? 0 : 8;   // this half-lane owns m = mbase..mbase+7
  const float4* wk2tab = (const float4*)(ws + WS_WK2);

  float att1v[NVV][4];
  v8f k2v[NVV];

#pragma unroll
  for (int v = 0; v < NVV; ++v) {
    // B operand: hidden^T for var v (K = channel, N = batch column = lane%16)
    v2f Bv1, Bv2;
    Bv1.x = lo ? hid[v * CVV + 0] : hid[v * CVV + 2];
    Bv1.y = lo ? hid[v * CVV + 1] : hid[v * CVV + 3];
    Bv2.x = lo ? hid[v * CVV + 4] : 0.f;
    Bv2.y = lo ? hid[v * CVV + 5] : 0.f;

    // att1[r, b] for this v : two chained K=4 WMMAs (K = 6 zero-padded to 8)
    v8f d = catt1;
    d = __builtin_amdgcn_wmma_f32_16x16x4_f32(false, A1a, false, Bv1,
                                              (short)0, d, false, false);
    d = __builtin_amdgcn_wmma_f32_16x16x4_f32(false, A1b, false, Bv2,
                                              (short)0, d, false, false);
    att1v[v][0] = d[0]; att1v[v][1] = d[1];
    att1v[v][2] = d[2]; att1v[v][3] = d[3];

    // k2[m, b] = wk2[v]^T @ hidden^T + bk2[v]  (bias via C operand)
    float4 a2 = wk2tab[v * 32 + lane];        // pre-swizzled A operand
    v2f A2a, A2b;
    A2a.x = a2.x; A2a.y = a2.y;
    A2b.x = a2.z; A2b.y = a2.w;
    const float4* bv = (const float4*)(bk2 + v * CM2C + mbase);
    float4 e0 = bv[0], e1 = bv[1];
    v8f e;
    e[0] = e0.x; e[1] = e0.y; e[2] = e0.z; e[3] = e0.w;
    e[4] = e1.x; e[5] = e1.y; e[6] = e1.z; e[7] = e1.w;
    e = __builtin_amdgcn_wmma_f32_16x16x4_f32(false, A2a, false, Bv1,
                                              (short)0, e, false, false);
    e = __builtin_amdgcn_wmma_f32_16x16x4_f32(false, A2b, false, Bv2,
                                              (short)0, e, false, false);
    k2v[v] = e;   // lane<16: m=j ; lane>=16: m=j+8
  }

  // ---- argmax over flat (r*NV+v), first max wins (lanes<16 hold real data)
  float best1 = -3.4e38f;
  int r_sel = 0, p_sel = 0;
#pragma unroll
  for (int r = 0; r < NRR; ++r)
#pragma unroll
    for (int v = 0; v < NVV; ++v) {
      float a = att1v[v][r];
      if (a > best1) { best1 = a; r_sel = r; p_sel = v; }
    }
  // broadcast selection to the upper half-lanes
  {
    int rs2 = __shfl_xor(r_sel, 16, 32);
    int ps2 = __shfl_xor(p_sel, 16, 32);
    r_sel = lo ? r_sel : rs2;
    p_sel = lo ? p_sel : ps2;
  }

  // ---- var_p = hidden[b, p_sel, :]  (constant-index selects, stays in VGPRs)
  float vp[CVV];
#pragma unroll
  for (int c = 0; c < CVV; ++c) {
    float x = hid[0 * CVV + c];
    if (p_sel == 1) x = hid[1 * CVV + c];
    if (p_sel == 2) x = hid[2 * CVV + c];
    if (p_sel == 3) x = hid[3 * CVV + c];
    if (p_sel == 4) x = hid[4 * CVV + c];
    if (p_sel == 5) x = hid[5 * CVV + c];
    vp[c] = x;
  }

  // ---- q2 (split: this half-lane computes q2[mbase .. mbase+7]) ----
  float q2p[8];
  {
    const float4* bq2v = (const float4*)bq2;
    float4 t0 = bq2v[mbase >> 2], t1 = bq2v[(mbase >> 2) + 1];
    q2p[0] = t0.x; q2p[1] = t0.y; q2p[2] = t0.z; q2p[3] = t0.w;
    q2p[4] = t1.x; q2p[5] = t1.y; q2p[6] = t1.z; q2p[7] = t1.w;
#pragma unroll
    for (int c = 0; c < CVV; ++c) {
      const float4* wv = (const float4*)(wq2 + c * CM2C + mbase);
      float4 wA = wv[0], wB = wv[1];
      q2p[0] = fmaf(vp[c], wA.x, q2p[0]); q2p[1] = fmaf(vp[c], wA.y, q2p[1]);
      q2p[2] = fmaf(vp[c], wA.z, q2p[2]); q2p[3] = fmaf(vp[c], wA.w, q2p[3]);
      q2p[4] = fmaf(vp[c], wB.x, q2p[4]); q2p[5] = fmaf(vp[c], wB.y, q2p[5]);
      q2p[6] = fmaf(vp[c], wB.z, q2p[6]); q2p[7] = fmaf(vp[c], wB.w, q2p[7]);
    }
  }

  // ---- att2[v] = (q2 . k2[v]) / 4 : partner-half partial sums combined ----
  float att2v[NVV];
#pragma unroll
  for (int v = 0; v < NVV; ++v) {
    float s = 0.f;
#pragma unroll
    for (int j = 0; j < 8; ++j) s = fmaf(q2p[j], k2v[v][j], s);
    s += __shfl_xor(s, 16, 32);
    att2v[v] = s * 0.25f; // 1/sqrt(16)
  }

  float best2 = -3.4e38f;
  int c_sel = 0;
#pragma unroll
  for (int v = 0; v < NVV; ++v)
    if (att2v[v] > best2) { best2 = att2v[v]; c_sel = v; }

  // var_pc = [vp[0], vp[1], var_c[0], var_c[1]]
  float x0 = vp[0], x1 = vp[1];
  float x2 = hid[0], x3 = hid[1];
#pragma unroll
  for (int v = 1; v < NVV; ++v) {
    if (c_sel == v) { x2 = hid[v * CVV + 0]; x3 = hid[v * CVV + 1]; }
  }

  // ---- rule MLP, selected rule only: relu(x @ w1[r] + b1[r]) @ w2[r] + b2[r]
  const float4* W1r = (const float4*)(w1 + (size_t)r_sel * 4 * 32);
  const float4* b1r = (const float4*)(b1 + (size_t)r_sel * 32);
  const float4* w2r = (const float4*)(w2 + (size_t)r_sel * 64);
  float o0 = b2[r_sel * 2 + 0], o1 = b2[r_sel * 2 + 1];
#pragma unroll
  for (int o4 = 0; o4 < 8; ++o4) {
    float4 hb  = b1r[o4];
    float4 wd0 = W1r[0 * 8 + o4], wd1 = W1r[1 * 8 + o4];
    float4 wd2 = W1r[2 * 8 + o4], wd3 = W1r[3 * 8 + o4];
    float h0 = fmaf(x3, wd3.x, fmaf(x2, wd2.x, fmaf(x1, wd1.x, fmaf(x0, wd0.x, hb.x))));
    float h1 = fmaf(x3, wd3.y, fmaf(x2, wd2.y, fmaf(x1, wd1.y, fmaf(x0, wd0.y, hb.y))));
    float h2 = fmaf(x3, wd3.z, fmaf(x2, wd2.z, fmaf(x1, wd1.z, fmaf(x0, wd0.z, hb.z))));
    float h3 = fmaf(x3, wd3.w, fmaf(x2, wd2.w, fmaf(x1, wd1.w, fmaf(x0, wd0.w, hb.w))));
    h0 = fmaxf(h0, 0.f); h1 = fmaxf(h1, 0.f);
    h2 = fmaxf(h2, 0.f); h3 = fmaxf(h3, 0.f);
    float4 wa = w2r[o4 * 2 + 0], wb = w2r[o4 * 2 + 1];
    o0 = fmaf(h3, wb.z, fmaf(h2, wb.x, fmaf(h1, wa.z, fmaf(h0, wa.x, o0))));
    o1 = fmaf(h3, wb.w, fmaf(h2, wb.y, fmaf(h1, wa.w, fmaf(h0, wa.y, o1))));
  }

  // ---- stores: output [B,6,2] then mask_p [B,6,1], zeros elsewhere ----
  if (lo) {
    float4 O0, O1, O2;
    O0.x = (p_sel == 0) ? o0 : 0.f; O0.y = (p_sel == 0) ? o1 : 0.f;
    O0.z = (p_sel == 1) ? o0 : 0.f; O0.w = (p_sel == 1) ? o1 : 0.f;
    O1.x = (p_sel == 2) ? o0 : 0.f; O1.y = (p_sel == 2) ? o1 : 0.f;
    O1.z = (p_sel == 3) ? o0 : 0.f; O1.w = (p_sel == 3) ? o1 : 0.f;
    O2.x = (p_sel == 4) ? o0 : 0.f; O2.y = (p_sel == 4) ? o1 : 0.f;
    O2.z = (p_sel == 5) ? o0 : 0.f; O2.w = (p_sel == 5) ? o1 : 0.f;
    float4* op = (float4*)(out + (size_t)b * 12);
    op[0] = O0; op[1] = O1; op[2] = O2;

    float2 M0, M1, M2;
    M0.x = (p_sel == 0) ? 1.f : 0.f; M0.y = (p_sel == 1) ? 1.f : 0.f;
    M1.x = (p_sel == 2) ? 1.f : 0.f; M1.y = (p_sel == 3) ? 1.f : 0.f;
    M2.x = (p_sel == 4) ? 1.f : 0.f; M2.y = (p_sel == 5) ? 1.f : 0.f;
    float2* mp = (float2*)(out + (size_t)batch * 12 + (size_t)b * 6);
    mp[0] = M0; mp[1] = M1; mp[2] = M2;
  }
}

extern "C" void kernel_launch(void* const* d_in, const int* in_sizes, int n_in,
                              void* d_out, int out_size, void* d_ws, size_t ws_size,
                              hipStream_t stream) {
  const float* hidden   = (const float*)d_in[0];
  const float* rule_emb = (const float*)d_in[1];
  const float* wq1 = (const float*)d_in[2];
  const float* bq1 = (const float*)d_in[3];
  const float* wk1 = (const float*)d_in[4];
  const float* bk1 = (const float*)d_in[5];
  const float* wq2 = (const float*)d_in[6];
  const float* bq2 = (const float*)d_in[7];
  const float* wk2 = (const float*)d_in[8];
  const float* bk2 = (const float*)d_in[9];
  const float* w1  = (const float*)d_in[10];
  const float* b1  = (const float*)d_in[11];
  const float* w2  = (const float*)d_in[12];
  const float* b2  = (const float*)d_in[13];
  float* ws  = (float*)d_ws;
  float* out = (float*)d_out;

  int batch = in_sizes[0] / (NVV * CVV);

  rulenet_prep<<<1, 256, 0, stream>>>(rule_emb, wq1, bq1, wk1, bk1, wk2, ws);

  int waves  = (batch + 15) / 16;
  int blocks = (waves + 7) / 8;   // 256 threads = 8 wave32 per block
  rulenet_main<<<blocks, 256, 0, stream>>>(hidden, wq2, bq2, bk2,
                                           w1, b1, w2, b2, ws, out, batch);
}